// local_transformer_agg_43920335569208
// MI455X (gfx1250) — compile-verified
//
#include <hip/hip_runtime.h>

// Problem constants (from reference)
#define BB   8
#define CC   128
#define NN   2000
#define KK   16
#define MPB  (NN * KK)       // 32000 columns per batch image
#define NBLK 4               // n-positions per workgroup
#define MCH  (NBLK * KK)     // 64 columns per chunk
#define RPSTR 160            // LDS row-pair stride in floats (128 data + 32 pad -> conflict-free b64 reads)
#define CHUNKS (NN / NBLK)   // 500 chunks per batch
#define IN_EPS 1e-3f
#define BN_EPS 1e-5f

typedef __attribute__((ext_vector_type(2))) float v2f;
typedef __attribute__((ext_vector_type(8))) float v8f;

// ws layout (floats): [0,1024) sum_bc  [1024,2048) sumsq_bc  [2048,3072) alpha_bc  [3072,4096) beta_bc

// DS_SWIZZLE xor-mode butterfly (group-of-32, and_mask=0x1F): one LDS-permute op per step,
// no shfl bounds-check VALU overhead. Masks <=8 keep lanes inside their 16-lane row group.
#define SWZ_XOR(x, m) __int_as_float(__builtin_amdgcn_ds_swizzle(__float_as_int(x), (0x1F | ((m) << 10))))

__device__ __forceinline__ float redmax16(float x) {
    x = fmaxf(x, SWZ_XOR(x, 1));
    x = fmaxf(x, SWZ_XOR(x, 2));
    x = fmaxf(x, SWZ_XOR(x, 4));
    x = fmaxf(x, SWZ_XOR(x, 8));
    return x;
}
__device__ __forceinline__ float redsum16(float x) {
    x += SWZ_XOR(x, 1);
    x += SWZ_XOR(x, 2);
    x += SWZ_XOR(x, 4);
    x += SWZ_XOR(x, 8);
    return x;
}

// Stage a 128-row x 64-col f32 tile into LDS, row-pair interleaved:
// element (r,c) lives at (r>>1)*RPSTR + 2c + (r&1), so a B fragment
// {X[r][c], X[r+1][c]} is one aligned ds_load_b64.
__device__ __forceinline__ void stage_tile(float* xs, const float* xb, int tid) {
    for (int i = tid; i < 64 * 32; i += 256) {      // 64 row-pairs x 32 col-pairs
        int rp = i >> 5;
        int cp = i & 31;
        const float* p0 = xb + (size_t)(2 * rp) * MPB + 2 * cp;
        v2f g0 = *reinterpret_cast<const v2f*>(p0);
        v2f g1 = *reinterpret_cast<const v2f*>(p0 + MPB);
        float4 w;
        w.x = g0.x; w.y = g1.x; w.z = g0.y; w.w = g1.y;
        *reinterpret_cast<float4*>(&xs[rp * RPSTR + 4 * cp]) = w;
    }
}

__global__ void k_zero(float* ws) {
    int i = blockIdx.x * 256 + threadIdx.x;
    if (i < 4 * BB * CC) ws[i] = 0.0f;
}

// ---------------------------------------------------------------------------
// Pass 1: h = W0 * x (raw, no bias -- bias cancels in the fused norm affine),
// accumulate per-(b,c) sum and sum-of-squares via f32 global atomics.
// ---------------------------------------------------------------------------
__global__ __launch_bounds__(256) void k_pass1(const float* __restrict__ xl,
                                               const float* __restrict__ w0,
                                               float* __restrict__ sums) {
    __shared__ float xs[64 * RPSTR];
    const int b    = blockIdx.x / CHUNKS;
    const int ch   = blockIdx.x % CHUNKS;
    const int m0   = ch * MCH;
    const int tid  = threadIdx.x;
    const int wave = tid >> 5;
    const int lane = tid & 31;
    const int hi   = lane >> 4;   // which 16-lane half
    const int lr   = lane & 15;
    const int cobase = wave * 16; // 8 waves x 16 rows = 128 output channels

    stage_tile(xs, xl + (size_t)b * CC * MPB + m0, tid);
    __syncthreads();

    const v8f zero = {0.f, 0.f, 0.f, 0.f, 0.f, 0.f, 0.f, 0.f};
    v8f acc[NBLK];
#pragma unroll
    for (int t = 0; t < NBLK; ++t) acc[t] = zero;

    for (int kb = 0; kb < CC; kb += 4) {
        // A fragment: lane holds W0[cobase+lr][kb+2hi], [kb+2hi+1] -- one global b64
        v2f a = *reinterpret_cast<const v2f*>(w0 + (cobase + lr) * CC + kb + 2 * hi);
        const float* brow = &xs[((kb >> 1) + hi) * RPSTR];
#pragma unroll
        for (int t = 0; t < NBLK; ++t) {
            v2f bf = *reinterpret_cast<const v2f*>(brow + 2 * (t * KK + lr));
            acc[t] = __builtin_amdgcn_wmma_f32_16x16x4_f32(
                false, a, false, bf, (short)0, acc[t], false, false);
        }
    }

    // Per-channel statistics: rows of D live on 16-lane groups
#pragma unroll
    for (int v = 0; v < 8; ++v) {
        float s = 0.f, sq = 0.f;
#pragma unroll
        for (int t = 0; t < NBLK; ++t) { float x = acc[t][v]; s += x; sq += x * x; }
        s  = redsum16(s);
        sq = redsum16(sq);
        if (lr == 0) {
            int co = cobase + v + (hi << 3);
            atomicAdd(&sums[b * CC + co], s);
            atomicAdd(&sums[BB * CC + b * CC + co], sq);
        }
    }
}

// ---------------------------------------------------------------------------
// Fold IN(eps=1e-3) -> BN(train stats, eps=1e-5) -> affine into per-(b,c)
// alpha/beta. BN mean is exactly 0; BN var = mean_b[ var/(var+eps_in) ].
// ---------------------------------------------------------------------------
__global__ void k_params(const float* __restrict__ sums,
                         const float* __restrict__ bng,
                         const float* __restrict__ bnb,
                         float* __restrict__ ab) {
    int c = threadIdx.x;
    if (c >= CC) return;
    const float invM = 1.0f / (float)MPB;
    float mean[BB], var[BB];
    float bacc = 0.f;
    for (int b = 0; b < BB; ++b) {
        float s  = sums[b * CC + c];
        float sq = sums[BB * CC + b * CC + c];
        float m  = s * invM;
        float v  = fmaxf(sq * invM - m * m, 0.f);
        mean[b] = m; var[b] = v;
        bacc += v / (v + IN_EPS);
    }
    float brs = rsqrtf(bacc * (1.0f / BB) + BN_EPS);
    float g = bng[c], bbv = bnb[c];
    for (int b = 0; b < BB; ++b) {
        float al = rsqrtf(var[b] + IN_EPS) * brs * g;
        ab[b * CC + c]           = al;                    // alpha (applies to raw W0*x)
        ab[BB * CC + b * CC + c] = bbv - mean[b] * al;    // beta  (b0 cancels)
    }
}

// ---------------------------------------------------------------------------
// Pass 2: recompute h, apply fused norm affine + ReLU (into same LDS buffer),
// q/k/v WMMA GEMMs, softmax over the 16 neighbors (16-lane group), residual.
// ---------------------------------------------------------------------------
__global__ __launch_bounds__(256) void k_pass2(const float* __restrict__ xr,
                                               const float* __restrict__ xl,
                                               const float* __restrict__ w0,
                                               const float* __restrict__ wq,
                                               const float* __restrict__ bq,
                                               const float* __restrict__ wk,
                                               const float* __restrict__ bk,
                                               const float* __restrict__ wv,
                                               const float* __restrict__ bv,
                                               const float* __restrict__ ab,
                                               const float* __restrict__ gamma1,
                                               float* __restrict__ out) {
    __shared__ float xs[64 * RPSTR];
    const int b    = blockIdx.x / CHUNKS;
    const int ch   = blockIdx.x % CHUNKS;
    const int m0   = ch * MCH;
    const int tid  = threadIdx.x;
    const int wave = tid >> 5;
    const int lane = tid & 31;
    const int hi   = lane >> 4;
    const int lr   = lane & 15;
    const int cobase = wave * 16;

    stage_tile(xs, xl + (size_t)b * CC * MPB + m0, tid);
    __syncthreads();

    const v8f zero = {0.f, 0.f, 0.f, 0.f, 0.f, 0.f, 0.f, 0.f};
    v8f acc[NBLK];
#pragma unroll
    for (int t = 0; t < NBLK; ++t) acc[t] = zero;

    for (int kb = 0; kb < CC; kb += 4) {
        v2f a = *reinterpret_cast<const v2f*>(w0 + (cobase + lr) * CC + kb + 2 * hi);
        const float* brow = &xs[((kb >> 1) + hi) * RPSTR];
#pragma unroll
        for (int t = 0; t < NBLK; ++t) {
            v2f bf = *reinterpret_cast<const v2f*>(brow + 2 * (t * KK + lr));
            acc[t] = __builtin_amdgcn_wmma_f32_16x16x4_f32(
                false, a, false, bf, (short)0, acc[t], false, false);
        }
    }

    // per-row fused norm params (broadcast loads within each 16-lane group)
    float al[8], be[8];
#pragma unroll
    for (int v = 0; v < 8; ++v) {
        int co = cobase + v + (hi << 3);
        al[v] = ab[b * CC + co];
        be[v] = ab[BB * CC + b * CC + co];
    }

    __syncthreads();   // everyone done reading xs; reuse buffer for h_relu
#pragma unroll
    for (int t = 0; t < NBLK; ++t)
#pragma unroll
        for (int v = 0; v < 8; ++v) {
            int co = cobase + v + (hi << 3);
            xs[(co >> 1) * RPSTR + 2 * (t * KK + lr) + (co & 1)] =
                fmaxf(al[v] * acc[t][v] + be[v], 0.f);
        }
    __syncthreads();

    const float g1 = gamma1[0];

    for (int t = 0; t < NBLK; ++t) {
        v8f qa = zero, ka = zero, va = zero;
        for (int kb = 0; kb < CC; kb += 4) {
            const float* brow = &xs[((kb >> 1) + hi) * RPSTR];
            v2f bf = *reinterpret_cast<const v2f*>(brow + 2 * (t * KK + lr));
            int wrow = (cobase + lr) * CC + kb + 2 * hi;
            v2f aq  = *reinterpret_cast<const v2f*>(wq + wrow);
            v2f akf = *reinterpret_cast<const v2f*>(wk + wrow);
            v2f avf = *reinterpret_cast<const v2f*>(wv + wrow);
            qa = __builtin_amdgcn_wmma_f32_16x16x4_f32(false, aq,  false, bf, (short)0, qa, false, false);
            ka = __builtin_amdgcn_wmma_f32_16x16x4_f32(false, akf, false, bf, (short)0, ka, false, false);
            va = __builtin_amdgcn_wmma_f32_16x16x4_f32(false, avf, false, bf, (short)0, va, false, false);
        }

        // softmax over the 16 neighbor columns (one (co,n) row per 16-lane group)
#pragma unroll
        for (int v = 0; v < 8; ++v) {
            int co = cobase + v + (hi << 3);
            float qv = qa[v] + bq[co];
            float kv = ka[v] + bk[co];
            float vv = va[v] + bv[co];
            float s  = qv * kv;
            float mx = redmax16(s);
            float p  = __expf(s - mx);
            float ps = redsum16(p);
            float pv = redsum16(p * vv);
            if (lr == 0) {
                size_t idx = ((size_t)b * CC + co) * NN + (size_t)(ch * NBLK + t);
                out[idx] = xr[idx] + g1 * (pv / ps);
            }
        }
    }
}

extern "C" void kernel_launch(void* const* d_in, const int* in_sizes, int n_in,
                              void* d_out, int out_size, void* d_ws, size_t ws_size,
                              hipStream_t stream) {
    const float* x_row   = (const float*)d_in[0];
    const float* x_local = (const float*)d_in[1];
    const float* w0      = (const float*)d_in[2];
    const float* b0      = (const float*)d_in[3];  (void)b0; // cancels analytically
    const float* bn_g    = (const float*)d_in[4];
    const float* bn_b    = (const float*)d_in[5];
    const float* wq      = (const float*)d_in[6];
    const float* bq      = (const float*)d_in[7];
    const float* wk      = (const float*)d_in[8];
    const float* bk      = (const float*)d_in[9];
    const float* wv      = (const float*)d_in[10];
    const float* bv      = (const float*)d_in[11];
    const float* gamma1  = (const float*)d_in[12];
    (void)in_sizes; (void)n_in; (void)out_size; (void)ws_size;

    float* ws   = (float*)d_ws;
    float* sums = ws;               // 2*B*C floats
    float* ab   = ws + 2 * BB * CC; // 2*B*C floats

    k_zero<<<dim3((4 * BB * CC + 255) / 256), dim3(256), 0, stream>>>(ws);

    dim3 grid(BB * CHUNKS);  // 4000 blocks
    k_pass1<<<grid, dim3(256), 0, stream>>>(x_local, w0, sums);
    k_params<<<dim3(1), dim3(128), 0, stream>>>(sums, bn_g, bn_b, ab);
    k_pass2<<<grid, dim3(256), 0, stream>>>(x_row, x_local, w0, wq, bq, wk, bk,
                                            wv, bv, ab, gamma1, (float*)d_out);
}